// UsersEncoder_79903571575232
// MI455X (gfx1250) — compile-verified
//
#include <hip/hip_runtime.h>

// Segment-mean for sorted segment ids (fused to_dense_batch().sum(1)/len):
//   out[s, 0:64] = mean of x rows whose batch id == s ; out[s, 64] = 0
// Phase 1: binary-search run boundaries (sorted ids -> contiguous runs).
// Phase 2: one wave32 per segment streams its contiguous rows and reduces the
//          64 columns through V_WMMA_F32_16X16X4_F32 with an all-ones A matrix
//          (D[m][n] = sum_k B[k][n]) -- exact f32, atomic-free, coalesced.

typedef __attribute__((ext_vector_type(2))) float v2f;
typedef __attribute__((ext_vector_type(8))) float v8f;

#define SEG_D 64

__global__ void seg_bounds_kernel(const int* __restrict__ batch, int T, int B,
                                  int* __restrict__ start) {
    int s = blockIdx.x * blockDim.x + threadIdx.x;
    if (s > B) return;
    if (s == B) { start[B] = T; return; }
    // lower_bound: first index i with batch[i] >= s
    int lo = 0, hi = T;
    while (lo < hi) {
        int mid = (lo + hi) >> 1;
        if (batch[mid] < s) lo = mid + 1; else hi = mid;
    }
    start[s] = lo;
}

__global__ __launch_bounds__(32) void seg_mean_wmma_kernel(
    const float* __restrict__ x, const int* __restrict__ start,
    float* __restrict__ out) {
    const int s    = blockIdx.x;
    const int lane = threadIdx.x;        // wave32: 0..31
    const int col  = lane & 15;          // N index within a 16-col group
    const int half = lane >> 4;          // 0 -> rows {r,r+1}, 1 -> rows {r+2,r+3}

    const int i0  = start[s];
    const int i1  = start[s + 1];
    const int cnt = i1 - i0;

    v8f c0 = {}, c1 = {}, c2 = {}, c3 = {};
    const v2f a = {1.0f, 1.0f};          // all-ones A => D rows = column sums

    const float* __restrict__ xb = x + (long long)i0 * SEG_D;

    int r = 0;
    for (; r + 4 <= cnt; r += 4) {
        const float* p0 = xb + (long long)(r + 2 * half) * SEG_D;
        const float* p1 = p0 + SEG_D;
        v2f b0 = { p0[col +  0], p1[col +  0] };
        v2f b1 = { p0[col + 16], p1[col + 16] };
        v2f b2 = { p0[col + 32], p1[col + 32] };
        v2f b3 = { p0[col + 48], p1[col + 48] };
        c0 = __builtin_amdgcn_wmma_f32_16x16x4_f32(false, a, false, b0, (short)0, c0, false, false);
        c1 = __builtin_amdgcn_wmma_f32_16x16x4_f32(false, a, false, b1, (short)0, c1, false, false);
        c2 = __builtin_amdgcn_wmma_f32_16x16x4_f32(false, a, false, b2, (short)0, c2, false, false);
        c3 = __builtin_amdgcn_wmma_f32_16x16x4_f32(false, a, false, b3, (short)0, c3, false, false);
    }
    if (r < cnt) {
        // Tail: zero-pad missing rows; branches reconverge before the WMMAs,
        // so EXEC is all-ones when they issue.
        const int r0 = r + 2 * half;
        const int r1 = r0 + 1;
        const float* p0 = xb + (long long)r0 * SEG_D;
        const float* p1 = p0 + SEG_D;
        v2f b0 = {0.f, 0.f}, b1 = {0.f, 0.f}, b2 = {0.f, 0.f}, b3 = {0.f, 0.f};
        if (r0 < cnt) { b0.x = p0[col]; b1.x = p0[col + 16]; b2.x = p0[col + 32]; b3.x = p0[col + 48]; }
        if (r1 < cnt) { b0.y = p1[col]; b1.y = p1[col + 16]; b2.y = p1[col + 32]; b3.y = p1[col + 48]; }
        c0 = __builtin_amdgcn_wmma_f32_16x16x4_f32(false, a, false, b0, (short)0, c0, false, false);
        c1 = __builtin_amdgcn_wmma_f32_16x16x4_f32(false, a, false, b1, (short)0, c1, false, false);
        c2 = __builtin_amdgcn_wmma_f32_16x16x4_f32(false, a, false, b2, (short)0, c2, false, false);
        c3 = __builtin_amdgcn_wmma_f32_16x16x4_f32(false, a, false, b3, (short)0, c3, false, false);
    }

    // C/D layout: VGPR0 on lanes 0-15 is (M=0, N=lane); A==ones makes every M
    // row identical, so lanes 0-15 hold the column sums directly.
    const float inv = 1.0f / (float)(cnt > 0 ? cnt : 1);
    if (lane < 16) {
        size_t base = (size_t)s * (SEG_D + 1);
        out[base + col +  0] = c0[0] * inv;
        out[base + col + 16] = c1[0] * inv;
        out[base + col + 32] = c2[0] * inv;
        out[base + col + 48] = c3[0] * inv;
        if (lane == 0) out[base + SEG_D] = 0.0f;
    }
}

extern "C" void kernel_launch(void* const* d_in, const int* in_sizes, int n_in,
                              void* d_out, int out_size, void* d_ws, size_t ws_size,
                              hipStream_t stream) {
    const float* x     = (const float*)d_in[0];
    const int*   batch = (const int*)d_in[1];   // sorted segment ids
    const int T = in_sizes[1];
    const int B = out_size / (SEG_D + 1);       // out is [B, 65]

    int* start = (int*)d_ws;                    // (B+1) ints, rewritten every call

    int nb = (B + 1 + 255) / 256;
    seg_bounds_kernel<<<nb, 256, 0, stream>>>(batch, T, B, start);
    seg_mean_wmma_kernel<<<B, 32, 0, stream>>>(x, start, (float*)d_out);
}